// CouncilCrossAttentionAdapter_32899449487398
// MI455X (gfx1250) — compile-verified
//
#include <hip/hip_runtime.h>
#include <hip/hip_bf16.h>
#include <math.h>

// Problem constants (from reference): B=4, T=4096, L=4096, H=1024
#define BB 4
#define TT 4096
#define LL 4096
#define HH 1024

// GEMM tiling
#define KC    128          // K-chunk staged in LDS per step
#define LDSW  132          // padded row stride (floats): bank advance 4/row -> conflict-free b64 reads
#define NCHUNK (HH / KC)   // 8

typedef __attribute__((ext_vector_type(2))) float v2f;
typedef __attribute__((ext_vector_type(8))) float v8f;

// ---------------------------------------------------------------------------
// GEMM: C[M,N] = (X[M,K] @ W[N,K]^T) * scale      (M=4096, N=K=H=1024)
//
// Per wave: 16(M) x 64(N) tile via 4 f32 WMMA accumulators (V_WMMA_F32_16X16X4_F32).
// Block (8 waves): 128 x 64 tile. Grid: (N/64, M/128).
//
// The 64xKC W slab is shared by all 8 waves -> staged once per block into LDS
// with GLOBAL_LOAD_ASYNC_TO_LDS_B128 (ASYNCcnt), double-buffered so the copy
// of chunk c+1 overlaps the 32 WMMA K-steps of chunk c. B fragments then come
// from ds_load_b64 (conflict-free via the 132-float row pad); A fragments are
// wave-private rows loaded directly from global (L2 resident).
// ---------------------------------------------------------------------------
__global__ __launch_bounds__(256)
void gemm_xwt_wmma(const float* __restrict__ X, const float* __restrict__ W,
                   float* __restrict__ C, const float* __restrict__ scale_ptr) {
    const int K = HH;
    const int N = HH;
    __shared__ __align__(16) float sw[2][64 * LDSW];

    const int tid  = threadIdx.x;
    const int lane = tid & 31;
    const int wave = tid >> 5;
    const int lr   = lane & 15;
    const int kp   = (lane >> 4) << 1;      // K-pair select: 0 or 2

    const int m0 = blockIdx.y * 128 + wave * 16;
    const int n0 = blockIdx.x * 64;

    const float* xrow  = X + (size_t)(m0 + lr) * K + kp;   // A fragment base
    const float* wtile = W + (size_t)n0 * K;               // uniform -> SGPR pair

    const uint32_t lds0 = (uint32_t)(uintptr_t)&sw[0][0];
    const uint32_t lds1 = (uint32_t)(uintptr_t)&sw[1][0];

    // Stage W[n0 .. n0+63][k0 .. k0+KC) into LDS buffer: 64*128 floats = 32KB,
    // 2048 16-byte pieces, 8 async b128 per thread. piece p = tid + 256*i,
    // r = p/32 (0..63), kk = (p%32)*4 (0..124).
    auto issue_chunk = [&](int k0, uint32_t ldsbase) {
#pragma unroll
        for (int i = 0; i < 8; ++i) {
            const int p  = tid + (i << 8);
            const int r  = p >> 5;
            const int kk = (p & 31) << 2;
            const uint32_t loff = ldsbase + (uint32_t)((r * LDSW + kk) << 2);
            const uint32_t goff = (uint32_t)((((size_t)r * K) + k0 + kk) << 2);
            asm volatile("global_load_async_to_lds_b128 %0, %1, %2"
                         :: "v"(loff), "v"(goff), "s"(wtile)
                         : "memory");
        }
    };

    v8f acc0 = {}; v8f acc1 = {}; v8f acc2 = {}; v8f acc3 = {};

    issue_chunk(0, lds0);

    for (int c = 0; c < NCHUNK; ++c) {
        if (c > 0) __syncthreads();   // all waves done reading buf[(c+1)&1] (chunk c-1)
        if (c + 1 < NCHUNK) {
            issue_chunk((c + 1) * KC, ((c + 1) & 1) ? lds1 : lds0);
            // chunk c's 8 per-wave async ops precede chunk c+1's 8: allow 8 outstanding
            asm volatile("s_wait_asynccnt 0x8" ::: "memory");
        } else {
            asm volatile("s_wait_asynccnt 0x0" ::: "memory");
        }
        __syncthreads();              // all waves' chunk-c data visible in LDS

        const float* xp = xrow + c * KC;
        const float* sb = &sw[c & 1][lr * LDSW + kp];
#pragma unroll 4
        for (int kk = 0; kk < KC; kk += 4) {
            v2f a  = *(const v2f*)(xp + kk);
            v2f b0 = *(const v2f*)(sb + kk);
            v2f b1 = *(const v2f*)(sb + 16 * LDSW + kk);
            v2f b2 = *(const v2f*)(sb + 32 * LDSW + kk);
            v2f b3 = *(const v2f*)(sb + 48 * LDSW + kk);
            acc0 = __builtin_amdgcn_wmma_f32_16x16x4_f32(false, a, false, b0, (short)0, acc0, false, false);
            acc1 = __builtin_amdgcn_wmma_f32_16x16x4_f32(false, a, false, b1, (short)0, acc1, false, false);
            acc2 = __builtin_amdgcn_wmma_f32_16x16x4_f32(false, a, false, b2, (short)0, acc2, false, false);
            acc3 = __builtin_amdgcn_wmma_f32_16x16x4_f32(false, a, false, b3, (short)0, acc3, false, false);
        }
    }

    const float scale = scale_ptr ? scale_ptr[0] : 1.0f;

    // D layout: VGPR r -> row (m0 + 8*(lane/16) + r), col n0 + j*16 + lane%16
    const int mrow = m0 + (lane >> 4) * 8;
    float* crow = C + (size_t)mrow * N + n0 + lr;
#pragma unroll
    for (int r = 0; r < 8; ++r) {
        crow[(size_t)r * N +  0] = acc0[r] * scale;
        crow[(size_t)r * N + 16] = acc1[r] * scale;
        crow[(size_t)r * N + 32] = acc2[r] * scale;
        crow[(size_t)r * N + 48] = acc3[r] * scale;
    }
}

// ---------------------------------------------------------------------------
// Per-token council attention. One wave32 per token t (8 tokens / block).
// Reads pre-projected qj/qi rows and gathered Kj/Vj/Ki/Vi rows, applies the
// logic-gate rewrite of score[0], two 3-way softmaxes, council mix, and
// writes the context row (aliased over the qj buffer).
// ---------------------------------------------------------------------------
__global__ __launch_bounds__(256)
void council_attn(const float* __restrict__ qj, const float* __restrict__ qi,
                  const float* __restrict__ Kj, const float* __restrict__ Vj,
                  const float* __restrict__ Ki, const float* __restrict__ Vi,
                  const int* __restrict__ adv_ids, const int* __restrict__ ptr_ids,
                  const float* __restrict__ council, float* __restrict__ ctx) {
    const int lane = threadIdx.x & 31;
    const int wave = threadIdx.x >> 5;
    const int t = blockIdx.x * 8 + wave;

    int p = ptr_ids[t];
    int i0 = min(max(p,     0), LL - 1);
    int i1 = min(max(p + 1, 0), LL - 1);
    int i2 = min(max(p + 2, 0), LL - 1);

    const size_t qoff = (size_t)t * HH;
    const float4* qjv = (const float4*)(qj + qoff);
    const float4* qiv = (const float4*)(qi + qoff);
    const float4* kj0 = (const float4*)(Kj + (size_t)i0 * HH);
    const float4* kj1 = (const float4*)(Kj + (size_t)i1 * HH);
    const float4* kj2 = (const float4*)(Kj + (size_t)i2 * HH);
    const float4* ki0 = (const float4*)(Ki + (size_t)i0 * HH);
    const float4* ki1 = (const float4*)(Ki + (size_t)i1 * HH);
    const float4* ki2 = (const float4*)(Ki + (size_t)i2 * HH);

    float sj0 = 0.f, sj1 = 0.f, sj2 = 0.f;
    float si0 = 0.f, si1 = 0.f, si2 = 0.f;

    const int nv = HH / 4;  // 256 float4 per row
    for (int it = lane; it < nv; it += 32) {
        float4 q = qjv[it];
        float4 k;
        k = kj0[it]; sj0 += q.x*k.x + q.y*k.y + q.z*k.z + q.w*k.w;
        k = kj1[it]; sj1 += q.x*k.x + q.y*k.y + q.z*k.z + q.w*k.w;
        k = kj2[it]; sj2 += q.x*k.x + q.y*k.y + q.z*k.z + q.w*k.w;
        float4 r = qiv[it];
        k = ki0[it]; si0 += r.x*k.x + r.y*k.y + r.z*k.z + r.w*k.w;
        k = ki1[it]; si1 += r.x*k.x + r.y*k.y + r.z*k.z + r.w*k.w;
        k = ki2[it]; si2 += r.x*k.x + r.y*k.y + r.z*k.z + r.w*k.w;
    }
    // wave32 xor reduction -> every lane holds the full sums
    for (int m = 16; m >= 1; m >>= 1) {
        sj0 += __shfl_xor(sj0, m); sj1 += __shfl_xor(sj1, m); sj2 += __shfl_xor(sj2, m);
        si0 += __shfl_xor(si0, m); si1 += __shfl_xor(si1, m); si2 += __shfl_xor(si2, m);
    }

    const float inv = 1.0f / sqrtf((float)HH);
    float lrel = sj0 * inv, l1 = sj1 * inv, l2 = sj2 * inv;

    const int rel = adv_ids[i0];          // wave-uniform
    float lfin;
    switch (rel) {
        case 0:  lfin = l1 + l2;            break;  // AND
        case 1:  lfin = fmaxf(l1, l2);      break;  // OR
        case 2:  lfin = -l1;                break;  // NOT
        case 3:  lfin = fmaxf(-l1, l2);     break;  // IMPLIES
        case 4:  lfin = fabsf(l1 - l2);     break;  // XOR
        default: lfin = lrel;               break;  // no gate
    }

    // softmax over {lfin, l1, l2}
    float mxj = fmaxf(lfin, fmaxf(l1, l2));
    float ej0 = __expf(lfin - mxj), ej1 = __expf(l1 - mxj), ej2 = __expf(l2 - mxj);
    float rsj = 1.0f / (ej0 + ej1 + ej2);
    float aj0 = ej0 * rsj, aj1 = ej1 * rsj, aj2 = ej2 * rsj;

    // softmax over si
    float t0 = si0 * inv, t1 = si1 * inv, t2 = si2 * inv;
    float mxi = fmaxf(t0, fmaxf(t1, t2));
    float ei0 = __expf(t0 - mxi), ei1 = __expf(t1 - mxi), ei2 = __expf(t2 - mxi);
    float rsi = 1.0f / (ei0 + ei1 + ei2);
    float ai0 = ei0 * rsi, ai1 = ei1 * rsi, ai2 = ei2 * rsi;

    // council softmax (2 elements)
    float c0 = council[0], c1 = council[1];
    float cm = fmaxf(c0, c1);
    float ec0 = __expf(c0 - cm), ec1 = __expf(c1 - cm);
    float rc = 1.0f / (ec0 + ec1);
    float w0 = ec0 * rc, w1 = ec1 * rc;

    const float fj0 = w0 * aj0, fj1 = w0 * aj1, fj2 = w0 * aj2;
    const float fi0 = w1 * ai0, fi1 = w1 * ai1, fi2 = w1 * ai2;

    const float4* vj0 = (const float4*)(Vj + (size_t)i0 * HH);
    const float4* vj1 = (const float4*)(Vj + (size_t)i1 * HH);
    const float4* vj2 = (const float4*)(Vj + (size_t)i2 * HH);
    const float4* vi0 = (const float4*)(Vi + (size_t)i0 * HH);
    const float4* vi1 = (const float4*)(Vi + (size_t)i1 * HH);
    const float4* vi2 = (const float4*)(Vi + (size_t)i2 * HH);
    float4* outr = (float4*)(ctx + qoff);

    for (int it = lane; it < nv; it += 32) {
        float4 v, o;
        v = vj0[it]; o.x = fj0*v.x; o.y = fj0*v.y; o.z = fj0*v.z; o.w = fj0*v.w;
        v = vj1[it]; o.x += fj1*v.x; o.y += fj1*v.y; o.z += fj1*v.z; o.w += fj1*v.w;
        v = vj2[it]; o.x += fj2*v.x; o.y += fj2*v.y; o.z += fj2*v.z; o.w += fj2*v.w;
        v = vi0[it]; o.x += fi0*v.x; o.y += fi0*v.y; o.z += fi0*v.z; o.w += fi0*v.w;
        v = vi1[it]; o.x += fi1*v.x; o.y += fi1*v.y; o.z += fi1*v.z; o.w += fi1*v.w;
        v = vi2[it]; o.x += fi2*v.x; o.y += fi2*v.y; o.z += fi2*v.z; o.w += fi2*v.w;
        outr[it] = o;
    }
}

// ---------------------------------------------------------------------------
// Launch: per batch b --
//   qj = hid_b @ Wqj^T ; qi = hid_b @ Wqi^T          (M = T)
//   Kj/Vj/Ki/Vi = adv_b @ W{kj,vj,ki,vi}^T            (M = L, PRE-gather!)
//   council_attn -> context row per token (aliased over qj)
//   out_b = context @ Wout^T * gain
// Workspace: 6 * T*H floats = 96 MB (context aliases qj). Per-batch K/V
// working set (64 MB) stays resident in the 192 MB L2 for the gather phase.
// ---------------------------------------------------------------------------
extern "C" void kernel_launch(void* const* d_in, const int* in_sizes, int n_in,
                              void* d_out, int out_size, void* d_ws, size_t ws_size,
                              hipStream_t stream) {
    const float* hidden  = (const float*)d_in[0];
    const float* advisor = (const float*)d_in[1];
    const int*   adv_ids = (const int*)  d_in[2];
    const int*   ptr_ids = (const int*)  d_in[3];
    const float* Wqj  = (const float*)d_in[4];
    const float* Wkj  = (const float*)d_in[5];
    const float* Wvj  = (const float*)d_in[6];
    const float* Wqi  = (const float*)d_in[7];
    const float* Wki  = (const float*)d_in[8];
    const float* Wvi  = (const float*)d_in[9];
    const float* Wout = (const float*)d_in[10];
    const float* gain    = (const float*)d_in[11];
    const float* council = (const float*)d_in[12];
    float* out = (float*)d_out;

    float* ws = (float*)d_ws;
    const size_t per = (size_t)TT * HH;
    float* qj = ws + 0 * per;
    float* qi = ws + 1 * per;
    float* Kj = ws + 2 * per;
    float* Vj = ws + 3 * per;
    float* Ki = ws + 4 * per;
    float* Vi = ws + 5 * per;
    float* ctx = qj;  // safe alias: attn reads qj row before writing ctx row

    const dim3 ggrid(HH / 64, TT / 128);
    const dim3 blk(256);

    for (int b = 0; b < BB; ++b) {
        const float* hb = hidden  + (size_t)b * TT * HH;
        const float* ab = advisor + (size_t)b * LL * HH;

        gemm_xwt_wmma<<<ggrid, blk, 0, stream>>>(hb, Wqj, qj, nullptr);
        gemm_xwt_wmma<<<ggrid, blk, 0, stream>>>(hb, Wqi, qi, nullptr);
        gemm_xwt_wmma<<<ggrid, blk, 0, stream>>>(ab, Wkj, Kj, nullptr);
        gemm_xwt_wmma<<<ggrid, blk, 0, stream>>>(ab, Wvj, Vj, nullptr);
        gemm_xwt_wmma<<<ggrid, blk, 0, stream>>>(ab, Wki, Ki, nullptr);
        gemm_xwt_wmma<<<ggrid, blk, 0, stream>>>(ab, Wvi, Vi, nullptr);

        council_attn<<<TT / 8, blk, 0, stream>>>(qj, qi, Kj, Vj, Ki, Vi,
                                                 adv_ids + (size_t)b * LL,
                                                 ptr_ids + (size_t)b * TT,
                                                 council, ctx);

        gemm_xwt_wmma<<<ggrid, blk, 0, stream>>>(ctx, Wout,
                                                 out + (size_t)b * TT * HH, gain);
    }
}